// GATEncoder_88802743812266
// MI455X (gfx1250) — compile-verified
//
#include <hip/hip_runtime.h>

// ---------------------------------------------------------------------------
// GAT encoder for MI455X (gfx1250, wave32).
//  - GEMMs via V_WMMA_F32_16X16X4_F32 (full f32 precision; GEMM is ~6.5 GFLOP,
//    far from the bottleneck, so no need for low precision).
//  - Edge scatter-softmax via L2 atomics (all hot data fits in 192MB L2).
// ---------------------------------------------------------------------------

typedef __attribute__((ext_vector_type(2))) float v2f;
typedef __attribute__((ext_vector_type(8))) float v8f;

// ---------------------------------------------------------------------------
// f32 WMMA GEMM:  C[M,N] = A[M,K] * B[K,N]   (row-major, K % 4 == 0, N % 64 == 0)
// Block = 128 threads (4 waves). Each wave computes a 16x64 tile:
//   rows  m0 = blockIdx.x*64 + wave*16
//   cols  n0 = blockIdx.y*64
// A fragment (16x4 f32): lanes 0-15 hold rows, VGPR0/1 = K pair {0,1} (half 0)
// or {2,3} (half 1). B fragment (4x16): VGPR0 = rows k / k+2, VGPR1 = k+1 / k+3.
// Out-of-range rows are clamped on load (EXEC must stay all-ones for WMMA)
// and guarded on store.
// ---------------------------------------------------------------------------
__global__ void __launch_bounds__(128)
gemm_wmma_f32(const float* __restrict__ A, const float* __restrict__ B,
              float* __restrict__ C, int M, int N, int K)
{
    const int wave = threadIdx.x >> 5;
    const int lane = threadIdx.x & 31;
    const int half = lane >> 4;   // 0: K pair {0,1}, 1: K pair {2,3}
    const int lr   = lane & 15;

    const int m0 = blockIdx.x * 64 + wave * 16;
    const int n0 = blockIdx.y * 64;

    v8f acc0 = {}, acc1 = {}, acc2 = {}, acc3 = {};

    int rowA = m0 + lr;
    if (rowA >= M) rowA = M - 1;              // clamp, keep EXEC all-ones
    const float* __restrict__ Arow = A + (size_t)rowA * K;

#pragma unroll 4
    for (int k = 0; k < K; k += 4) {
        v2f a = *(const v2f*)(Arow + k + 2 * half);

        const float* __restrict__ Bk0 = B + (size_t)(k + 2 * half) * N + n0 + lr;
        const float* __restrict__ Bk1 = Bk0 + N;
        v2f b0, b1, b2, b3;
        b0.x = Bk0[0];  b0.y = Bk1[0];
        b1.x = Bk0[16]; b1.y = Bk1[16];
        b2.x = Bk0[32]; b2.y = Bk1[32];
        b3.x = Bk0[48]; b3.y = Bk1[48];

        acc0 = __builtin_amdgcn_wmma_f32_16x16x4_f32(false, a, false, b0, (short)0, acc0, false, false);
        acc1 = __builtin_amdgcn_wmma_f32_16x16x4_f32(false, a, false, b1, (short)0, acc1, false, false);
        acc2 = __builtin_amdgcn_wmma_f32_16x16x4_f32(false, a, false, b2, (short)0, acc2, false, false);
        acc3 = __builtin_amdgcn_wmma_f32_16x16x4_f32(false, a, false, b3, (short)0, acc3, false, false);
    }

    // C/D layout: VGPR i, lane (half,lr) -> row m0 + i + 8*half, col n0 + 16*j + lr
#pragma unroll
    for (int i = 0; i < 8; ++i) {
        int row = m0 + i + 8 * half;
        if (row < M) {
            float* Crow = C + (size_t)row * N + n0 + lr;
            Crow[0]  = acc0[i];
            Crow[16] = acc1[i];
            Crow[32] = acc2[i];
            Crow[48] = acc3[i];
        }
    }
}

// ---------------------------------------------------------------------------
// Per-node attention logits: a_src[n,h] = dot(h[n,h,:], att_s[h,:]), same a_dst
// ---------------------------------------------------------------------------
__global__ void att_logits(const float* __restrict__ feat,
                           const float* __restrict__ att_s,
                           const float* __restrict__ att_d,
                           int Nn, int H, int C,
                           float* __restrict__ a_s, float* __restrict__ a_d)
{
    int idx = blockIdx.x * blockDim.x + threadIdx.x;
    if (idx >= Nn * H) return;
    int n = idx / H, h = idx - n * H;
    const float* row = feat + (size_t)n * H * C + h * C;
    float ss = 0.f, dd = 0.f;
    for (int c = 0; c < C; ++c) {
        float v = row[c];
        ss += v * att_s[h * C + c];
        dd += v * att_d[h * C + c];
    }
    a_s[idx] = ss;
    a_d[idx] = dd;
}

// ---------------------------------------------------------------------------
// Buffer init helpers (graph-capture safe, deterministic)
// ---------------------------------------------------------------------------
__global__ void fill_f32(float* p, long long n, float v)
{
    long long i = (long long)blockIdx.x * blockDim.x + threadIdx.x;
    if (i < n) p[i] = v;
}
__global__ void fill_i32(int* p, long long n, int v)
{
    long long i = (long long)blockIdx.x * blockDim.x + threadIdx.x;
    if (i < n) p[i] = v;
}

// order-preserving float->int encode for atomic max
__device__ __forceinline__ int enc_f32(float f)
{
    int i = __float_as_int(f);
    return i >= 0 ? i : (i ^ 0x7FFFFFFF);
}
__device__ __forceinline__ float dec_f32(int i)
{
    return i >= 0 ? __int_as_float(i) : __int_as_float(i ^ 0x7FFFFFFF);
}

// ---------------------------------------------------------------------------
// Edge pass 1: m[dst,h] = max over edges of leaky_relu(a_src[src,h]+a_dst[dst,h])
// Edges e < E0 come from edge_index; e >= E0 are self loops (src=dst=e-E0).
// ---------------------------------------------------------------------------
__global__ void edge_max(const long long* __restrict__ ei, int E0, int Etot,
                         const float* __restrict__ a_src,
                         const float* __restrict__ a_dst,
                         int H, int* __restrict__ mkey)
{
    int e = blockIdx.x * blockDim.x + threadIdx.x;
    if (e >= Etot) return;
    int s, d;
    if (e < E0) { s = (int)ei[e]; d = (int)ei[E0 + e]; }
    else        { s = e - E0; d = s; }
    for (int h = 0; h < H; ++h) {
        float v = a_src[s * H + h] + a_dst[d * H + h];
        v = v > 0.f ? v : 0.2f * v;                    // leaky_relu(0.2)
        atomicMax(&mkey[d * H + h], enc_f32(v));
    }
}

// ---------------------------------------------------------------------------
// Edge pass 2 (one wave32 per edge):
//   ex = exp(leaky(e) - m[dst]); denom[dst,h] += ex; agg[dst,:] += ex*feat[src,:]
// Lane l owns F/32 contiguous features -> coalesced b128 loads; lane's head is
// uniform within its group. Non-returning f32 atomics execute in L2.
// ---------------------------------------------------------------------------
template <int H, int C>
__global__ void __launch_bounds__(256)
edge_aggregate(const long long* __restrict__ ei, int E0, int Etot,
               const float* __restrict__ a_src, const float* __restrict__ a_dst,
               const int* __restrict__ mkey,
               const float* __restrict__ feat,
               float* __restrict__ denom, float* __restrict__ agg)
{
    constexpr int F   = H * C;
    constexpr int FPL = F / 32;          // features per lane
    int wid  = (blockIdx.x * blockDim.x + threadIdx.x) >> 5;  // edge id (wave-uniform)
    if (wid >= Etot) return;
    int lane = threadIdx.x & 31;

    int s, d;
    if (wid < E0) { s = (int)ei[wid]; d = (int)ei[E0 + wid]; }
    else          { s = wid - E0; d = s; }

    const int f0 = lane * FPL;
    const int h  = f0 / C;

    float e = a_src[s * H + h] + a_dst[d * H + h];
    e = e > 0.f ? e : 0.2f * e;
    float m  = dec_f32(mkey[d * H + h]);
    float ex = __expf(e - m);

    if ((f0 & (C - 1)) == 0)                       // one lane per head
        atomicAdd(&denom[d * H + h], ex);

    const float* __restrict__ fs = feat + (size_t)s * F + f0;
    float* __restrict__ ag = agg + (size_t)d * F + f0;
#pragma unroll
    for (int j = 0; j < FPL; ++j)
        atomicAdd(&ag[j], ex * fs[j]);
}

// ---------------------------------------------------------------------------
// Node finalize, layer 1: act = elu(agg/denom + bias)   (in place)
// ---------------------------------------------------------------------------
__global__ void finalize_elu(float* __restrict__ agg,
                             const float* __restrict__ denom,
                             const float* __restrict__ bias,
                             int Nn, int H, int C)
{
    int F = H * C;
    long long i = (long long)blockIdx.x * blockDim.x + threadIdx.x;
    if (i >= (long long)Nn * F) return;
    int n = (int)(i / F);
    int f = (int)(i - (long long)n * F);
    int h = f / C;
    float x = agg[i] / denom[n * H + h] + bias[f];
    agg[i] = x > 0.f ? x : (__expf(x) - 1.0f);     // elu
}

// ---------------------------------------------------------------------------
// Node finalize, layer 2: out = out/denom + bias   (in place, no activation)
// ---------------------------------------------------------------------------
__global__ void finalize_out(float* __restrict__ out,
                             const float* __restrict__ denom,
                             const float* __restrict__ bias,
                             int Nn, int F)
{
    long long i = (long long)blockIdx.x * blockDim.x + threadIdx.x;
    if (i >= (long long)Nn * F) return;
    int n = (int)(i / F);
    int f = (int)(i - (long long)n * F);
    out[i] = out[i] / denom[n] + bias[f];
}

// ---------------------------------------------------------------------------
extern "C" void kernel_launch(void* const* d_in, const int* in_sizes, int n_in,
                              void* d_out, int out_size, void* d_ws, size_t ws_size,
                              hipStream_t stream)
{
    const float*      x      = (const float*)d_in[0];
    const long long*  ei     = (const long long*)d_in[1];   // int64 per reference
    const float*      W1     = (const float*)d_in[2];
    const float*      att_s1 = (const float*)d_in[3];
    const float*      att_d1 = (const float*)d_in[4];
    const float*      bias1  = (const float*)d_in[5];
    const float*      W2     = (const float*)d_in[6];
    const float*      att_s2 = (const float*)d_in[7];
    const float*      att_d2 = (const float*)d_in[8];
    const float*      bias2  = (const float*)d_in[9];

    const int F_in = 128, H1 = 4, C1 = 64, F1 = H1 * C1 /*256*/, F2 = 128;
    const int N    = in_sizes[0] / F_in;       // 50000
    const int E0   = in_sizes[1] / 2;          // 800000
    const int Etot = E0 + N;                   // with self loops

    // ---- workspace layout (floats) ----
    float* ws   = (float*)d_ws;
    float* h1   = ws;                       // N*256   (reused as h2: N*128)
    float* agg1 = h1   + (size_t)N * F1;    // N*256   (reused as ELU activations)
    float* a_s1 = agg1 + (size_t)N * F1;    // N*4
    float* a_d1 = a_s1 + (size_t)N * H1;    // N*4
    int*   m1   = (int*)(a_d1 + (size_t)N * H1);   // N*4
    float* den1 = (float*)(m1 + (size_t)N * H1);   // N*4
    float* a_s2 = den1 + (size_t)N * H1;    // N
    float* a_d2 = a_s2 + (size_t)N;         // N
    int*   m2   = (int*)(a_d2 + (size_t)N); // N
    float* den2 = (float*)(m2 + (size_t)N); // N
    float* h2   = h1;                       // reuse (layer-1 h no longer needed)
    float* out  = (float*)d_out;            // layer-2 aggregation target

    const int TB = 256;
    auto blocks = [](long long n, int tb) { return (unsigned)((n + tb - 1) / tb); };

    // ================= Layer 1 =================
    {
        dim3 grid((N + 63) / 64, F1 / 64);
        gemm_wmma_f32<<<grid, 128, 0, stream>>>(x, W1, h1, N, F1, F_in);
    }
    att_logits<<<blocks((long long)N * H1, TB), TB, 0, stream>>>(
        h1, att_s1, att_d1, N, H1, C1, a_s1, a_d1);

    fill_i32<<<blocks((long long)N * H1, TB), TB, 0, stream>>>(m1, (long long)N * H1, (int)0x80000000);
    fill_f32<<<blocks((long long)N * H1, TB), TB, 0, stream>>>(den1, (long long)N * H1, 0.f);
    fill_f32<<<blocks((long long)N * F1, TB), TB, 0, stream>>>(agg1, (long long)N * F1, 0.f);

    edge_max<<<blocks(Etot, TB), TB, 0, stream>>>(ei, E0, Etot, a_s1, a_d1, H1, m1);

    edge_aggregate<4, 64><<<blocks((long long)Etot * 32, TB), TB, 0, stream>>>(
        ei, E0, Etot, a_s1, a_d1, m1, h1, den1, agg1);

    finalize_elu<<<blocks((long long)N * F1, TB), TB, 0, stream>>>(
        agg1, den1, bias1, N, H1, C1);

    // ================= Layer 2 =================
    {
        dim3 grid((N + 63) / 64, F2 / 64);
        gemm_wmma_f32<<<grid, 128, 0, stream>>>(agg1, W2, h2, N, F2, F1);
    }
    att_logits<<<blocks(N, TB), TB, 0, stream>>>(
        h2, att_s2, att_d2, N, 1, F2, a_s2, a_d2);

    fill_i32<<<blocks(N, TB), TB, 0, stream>>>(m2, N, (int)0x80000000);
    fill_f32<<<blocks(N, TB), TB, 0, stream>>>(den2, N, 0.f);
    fill_f32<<<blocks((long long)N * F2, TB), TB, 0, stream>>>(out, (long long)N * F2, 0.f);

    edge_max<<<blocks(Etot, TB), TB, 0, stream>>>(ei, E0, Etot, a_s2, a_d2, 1, m2);

    edge_aggregate<1, 128><<<blocks((long long)Etot * 32, TB), TB, 0, stream>>>(
        ei, E0, Etot, a_s2, a_d2, m2, h2, den2, out);

    finalize_out<<<blocks((long long)N * F2, TB), TB, 0, stream>>>(
        out, den2, bias2, N, F2);
}